// NOVAE_34359738461
// MI455X (gfx1250) — compile-verified
//
#include <hip/hip_runtime.h>

// ---------------------------------------------------------------------------
// CDNA5 (gfx1250) NOVAE forward: encoder MLP -> cost matrix -> Sinkhorn ->
// soft coupling -> decoder MLP.
//  * GEMMs: V_WMMA_F32_16X16X4_F32, 32x32 tile per wave (2x2 accumulators),
//    64x64 per 4-wave block.
//  * Sinkhorn: K and K^T both resident in the 192MB L2; both half-steps are
//    block-per-row dots (4096 blocks) with float4 loads; the multiplier
//    vector is staged to LDS with gfx1250 async-to-LDS DMA.
// ---------------------------------------------------------------------------

typedef __attribute__((ext_vector_type(2))) float v2f;
typedef __attribute__((ext_vector_type(8))) float v8f;

#define BATCH   4096
#define NPRIOR  4096
#define DLAT    128
#define DIN     1024
#define SINK_REG 0.05f
#define SINK_ITERS 200
#define COL_SPLIT 16

// ---------------------------------------------------------------------------
// WMMA fp32 GEMM:  C[M,N] = A[M,K] @ B  (+bias, +relu)
//   TRANSB=false: B row-major [K,N];  TRANSB=true: B row-major [N,K]
// Wave tile 32x32 (acc00/01/10/11); block = 4 waves in 2x2 -> 64x64.
// Fragment layouts per CDNA5 ISA 7.12.2.
// ---------------------------------------------------------------------------
template<bool TRANSB, bool RELU, bool BIAS>
__global__ __launch_bounds__(128)
void wmma_gemm_f32(const float* __restrict__ A, const float* __restrict__ Bm,
                   const float* __restrict__ bias, float* __restrict__ C,
                   int M, int N, int K) {
  const int wid  = threadIdx.x >> 5;
  const int lane = threadIdx.x & 31;
  const int half = lane >> 4;        // lane group 0/1
  const int l    = lane & 15;
  const int row0 = blockIdx.y * 64 + (wid >> 1) * 32;
  const int col0 = blockIdx.x * 64 + (wid & 1) * 32;
  if (row0 >= M || col0 >= N) return;

  v8f acc00 = {}, acc01 = {}, acc10 = {}, acc11 = {};
  const float* Ar0 = A + (size_t)(row0 + l) * K;
  const float* Ar1 = Ar0 + (size_t)16 * K;
  const float* Bt0 = TRANSB ? (Bm + (size_t)(col0 + l) * K) : nullptr;
  const float* Bt1 = TRANSB ? (Bt0 + (size_t)16 * K) : nullptr;

  for (int kk = 0; kk < K; kk += 16) {
#pragma unroll
    for (int s = 0; s < 4; ++s) {
      const int k = kk + 4 * s + 2 * half;
      v2f a0 = *(const v2f*)(Ar0 + k);
      v2f a1 = *(const v2f*)(Ar1 + k);
      v2f b0, b1;
      if (TRANSB) {
        b0 = *(const v2f*)(Bt0 + k);
        b1 = *(const v2f*)(Bt1 + k);
      } else {
        const float* Bk0 = Bm + (size_t)k * N + col0 + l;
        const float* Bk1 = Bk0 + N;
        b0.x = Bk0[0];  b0.y = Bk1[0];
        b1.x = Bk0[16]; b1.y = Bk1[16];
      }
      acc00 = __builtin_amdgcn_wmma_f32_16x16x4_f32(false, a0, false, b0, (short)0, acc00, false, false);
      acc01 = __builtin_amdgcn_wmma_f32_16x16x4_f32(false, a0, false, b1, (short)0, acc01, false, false);
      acc10 = __builtin_amdgcn_wmma_f32_16x16x4_f32(false, a1, false, b0, (short)0, acc10, false, false);
      acc11 = __builtin_amdgcn_wmma_f32_16x16x4_f32(false, a1, false, b1, (short)0, acc11, false, false);
    }
  }

  float bv0 = 0.0f, bv1 = 0.0f;
  if (BIAS) { bv0 = bias[col0 + l]; bv1 = bias[col0 + 16 + l]; }
  float* Cr = C + (size_t)(row0 + 8 * half) * N + col0 + l;
#pragma unroll
  for (int t = 0; t < 8; ++t) {
    float e00 = acc00[t], e01 = acc01[t], e10 = acc10[t], e11 = acc11[t];
    if (BIAS) { e00 += bv0; e01 += bv1; e10 += bv0; e11 += bv1; }
    if (RELU) {
      e00 = fmaxf(e00, 0.f); e01 = fmaxf(e01, 0.f);
      e10 = fmaxf(e10, 0.f); e11 = fmaxf(e11, 0.f);
    }
    Cr[(size_t)t * N]            = e00;
    Cr[(size_t)t * N + 16]       = e01;
    Cr[(size_t)(t + 16) * N]     = e10;
    Cr[(size_t)(t + 16) * N + 16] = e11;
  }
}

// ---------------------------------------------------------------------------
// 32x32 tiled transpose (LDS, padded to kill bank conflicts): KT = K^T
// ---------------------------------------------------------------------------
__global__ __launch_bounds__(256)
void transpose_kernel(const float* __restrict__ in, float* __restrict__ outm, int N) {
  __shared__ float tile[32][33];
  const int bx = blockIdx.x * 32, by = blockIdx.y * 32;
  const int tx = threadIdx.x & 31, ty = threadIdx.x >> 5;
  for (int r = ty; r < 32; r += 8)
    tile[r][tx] = in[(size_t)(by + r) * N + bx + tx];
  __syncthreads();
  for (int r = ty; r < 32; r += 8)
    outm[(size_t)(bx + r) * N + by + tx] = tile[tx][r];
}

// ---------------------------------------------------------------------------
// out[i] = marg / dot(Mat[i,:], vec)  — block per row, float4 loads.
// vec (16KB) staged into LDS via gfx1250 async-to-LDS DMA (b128/lane).
// ---------------------------------------------------------------------------
__global__ __launch_bounds__(256)
void rowdot_kernel(const float* __restrict__ Mat, const float* __restrict__ vec,
                   float* __restrict__ outv, int N, float marg) {
  __shared__ float lvec[NPRIOR];
  const int t = threadIdx.x;
  for (int c = t; c < N / 4; c += 256) {
    unsigned lo = (unsigned)(uintptr_t)(&lvec[4 * c]);            // LDS byte offset
    unsigned long long ga = (unsigned long long)(uintptr_t)(vec + 4 * c);
    asm volatile("global_load_async_to_lds_b128 %0, %1, off"
                 :: "v"(lo), "v"(ga) : "memory");
  }
  asm volatile("s_wait_asynccnt 0x0" ::: "memory");
  __syncthreads();

  const int i = blockIdx.x;
  const float4* row = (const float4*)(Mat + (size_t)i * N);
  const float4* lv  = (const float4*)lvec;
  float acc = 0.0f;
  for (int j = t; j < N / 4; j += 256) {
    float4 a = row[j], b = lv[j];
    acc += a.x * b.x + a.y * b.y + a.z * b.z + a.w * b.w;
  }
  __shared__ float s[256];
  s[t] = acc;
  __syncthreads();
  for (int st = 128; st > 0; st >>= 1) {
    if (t < st) s[t] += s[t + st];
    __syncthreads();
  }
  if (t == 0) outv[i] = marg / s[0];
}

// ---------------------------------------------------------------------------
// Fallback (no K^T buffer): split column matvec into row chunks + reduce.
// ---------------------------------------------------------------------------
__global__ __launch_bounds__(256)
void sink_col_part(const float* __restrict__ Km, const float* __restrict__ u,
                   float* __restrict__ part, int B, int N) {
  const int j = blockIdx.x * 256 + threadIdx.x;
  if (j >= N) return;
  const int chunkRows = B / COL_SPLIT;
  const int i0 = blockIdx.y * chunkRows;
  const float* p = Km + (size_t)i0 * N + j;
  float a0 = 0.f, a1 = 0.f, a2 = 0.f, a3 = 0.f;
  for (int i = 0; i < chunkRows; i += 4) {
    a0 += p[(size_t)(i + 0) * N] * u[i0 + i + 0];
    a1 += p[(size_t)(i + 1) * N] * u[i0 + i + 1];
    a2 += p[(size_t)(i + 2) * N] * u[i0 + i + 2];
    a3 += p[(size_t)(i + 3) * N] * u[i0 + i + 3];
  }
  part[(size_t)blockIdx.y * N + j] = (a0 + a1) + (a2 + a3);
}

__global__ __launch_bounds__(256)
void sink_col_finish(const float* __restrict__ part, float* __restrict__ v,
                     int N, float marg) {
  const int j = blockIdx.x * 256 + threadIdx.x;
  if (j >= N) return;
  float acc = 0.0f;
  for (int c = 0; c < COL_SPLIT; ++c) acc += part[(size_t)c * N + j];
  v[j] = marg / acc;
}

// ---------------------------------------------------------------------------
__global__ __launch_bounds__(128)
void rownorm_kernel(const float* __restrict__ X, float* __restrict__ out, int D) {
  const int i = blockIdx.x;
  const float* row = X + (size_t)i * D;
  float acc = 0.0f;
  for (int j = threadIdx.x; j < D; j += 128) { float v = row[j]; acc += v * v; }
  __shared__ float s[128];
  s[threadIdx.x] = acc;
  __syncthreads();
  for (int st = 64; st > 0; st >>= 1) {
    if (threadIdx.x < st) s[threadIdx.x] += s[threadIdx.x + st];
    __syncthreads();
  }
  if (threadIdx.x == 0) out[i] = s[0];
}

__global__ __launch_bounds__(256)
void cost_max_kernel(float* __restrict__ Mat, const float* __restrict__ zz,
                     const float* __restrict__ pp, unsigned* __restrict__ maxbits,
                     int N) {
  const int i = blockIdx.y;
  const int j = blockIdx.x * 256 + threadIdx.x;
  if (j >= N) return;
  const size_t off = (size_t)i * N + j;
  float val = fmaxf(zz[i] + pp[j] - 2.0f * Mat[off], 0.0f);
  Mat[off] = val;
  __shared__ float s[256];
  s[threadIdx.x] = val;
  __syncthreads();
  for (int st = 128; st > 0; st >>= 1) {
    if (threadIdx.x < st) s[threadIdx.x] = fmaxf(s[threadIdx.x], s[threadIdx.x + st]);
    __syncthreads();
  }
  if (threadIdx.x == 0) atomicMax(maxbits, __float_as_uint(s[0]));
}

__global__ __launch_bounds__(256)
void expK_kernel(float* __restrict__ Mat, const unsigned* __restrict__ maxbits,
                 long long total) {
  const float mx = __uint_as_float(*maxbits);
  const float scale = -1.0f / (SINK_REG * (mx + 1e-12f));
  const long long idx = (long long)blockIdx.x * 256 + threadIdx.x;
  if (idx < total) Mat[idx] = expf(Mat[idx] * scale);
}

__global__ __launch_bounds__(256)
void scale_pi_kernel(float* __restrict__ Km, const float* __restrict__ u,
                     const float* __restrict__ v, int N) {
  const int i = blockIdx.y;
  const int j = blockIdx.x * 256 + threadIdx.x;
  if (j >= N) return;
  const size_t off = (size_t)i * N + j;
  Km[off] = u[i] * Km[off] * v[j];
}

__global__ __launch_bounds__(256)
void init_kernel(float* __restrict__ u, float* __restrict__ v,
                 unsigned* __restrict__ maxbits, int n, float val) {
  const int i = blockIdx.x * 256 + threadIdx.x;
  if (i < n) { u[i] = val; v[i] = val; }
  if (i == 0) *maxbits = 0u;
}

// ---------------------------------------------------------------------------
static void launch_gemm(const float* A, const float* B, const float* bias,
                        float* C, int M, int N, int K, bool relu, bool transb,
                        hipStream_t s) {
  dim3 grid(N / 64, M / 64), block(128);
  if (transb) {
    wmma_gemm_f32<true, false, false><<<grid, block, 0, s>>>(A, B, nullptr, C, M, N, K);
  } else if (bias && relu) {
    wmma_gemm_f32<false, true, true><<<grid, block, 0, s>>>(A, B, bias, C, M, N, K);
  } else if (bias) {
    wmma_gemm_f32<false, false, true><<<grid, block, 0, s>>>(A, B, bias, C, M, N, K);
  } else {
    wmma_gemm_f32<false, false, false><<<grid, block, 0, s>>>(A, B, nullptr, C, M, N, K);
  }
}

extern "C" void kernel_launch(void* const* d_in, const int* in_sizes, int n_in,
                              void* d_out, int out_size, void* d_ws, size_t ws_size,
                              hipStream_t stream) {
  (void)in_sizes; (void)n_in; (void)out_size;

  const float* x       = (const float*)d_in[0];
  const float* z_prior = (const float*)d_in[1];
  const float* ew[4] = {(const float*)d_in[2], (const float*)d_in[4],
                        (const float*)d_in[6], (const float*)d_in[8]};
  const float* eb[4] = {(const float*)d_in[3], (const float*)d_in[5],
                        (const float*)d_in[7], (const float*)d_in[9]};
  const float* dw[4] = {(const float*)d_in[10], (const float*)d_in[12],
                        (const float*)d_in[14], (const float*)d_in[16]};
  const float* db[4] = {(const float*)d_in[11], (const float*)d_in[13],
                        (const float*)d_in[15], (const float*)d_in[17]};
  float* out = (float*)d_out;

  // Workspace layout
  const size_t actElems = (size_t)BATCH * DIN;     // 4M floats
  const size_t kElems   = (size_t)BATCH * NPRIOR;  // 16.7M floats
  float* actA = (float*)d_ws;
  float* actB = actA + actElems;
  float* Kmat = actB + actElems;
  float* zz   = Kmat + kElems;
  float* pp   = zz + BATCH;
  float* u    = pp + NPRIOR;
  float* v    = u + BATCH;
  unsigned* maxbits = (unsigned*)(v + NPRIOR);
  float* aux = (float*)(maxbits + 16);             // KT (64MB) or partials (256KB)
  float* KT   = aux;
  float* part = aux;
  const size_t needKT = (size_t)((char*)(KT + kElems) - (char*)d_ws);
  const bool useKT = (ws_size >= needKT);          // constant per run -> deterministic

  init_kernel<<<dim3((BATCH + 255) / 256), 256, 0, stream>>>(u, v, maxbits, BATCH,
                                                             1.0f / (float)BATCH);

  // ---- Encoder: 1024 -> 512 -> 1024 -> 512 -> 128 ----
  launch_gemm(x,    ew[0], eb[0], actA, BATCH, 512,  1024, true,  false, stream);
  launch_gemm(actA, ew[1], eb[1], actB, BATCH, 1024, 512,  true,  false, stream);
  launch_gemm(actB, ew[2], eb[2], actA, BATCH, 512,  1024, true,  false, stream);
  launch_gemm(actA, ew[3], eb[3], actB, BATCH, DLAT, 512,  false, false, stream);
  // actB = z_ (4096 x 128)

  // ---- Cost matrix M = relu(|z|^2 + |p|^2 - 2 z p^T), then K = exp(-M/(reg*max)) ----
  rownorm_kernel<<<dim3(BATCH), 128, 0, stream>>>(actB, zz, DLAT);
  rownorm_kernel<<<dim3(NPRIOR), 128, 0, stream>>>(z_prior, pp, DLAT);
  launch_gemm(actB, z_prior, nullptr, Kmat, BATCH, NPRIOR, DLAT, false, true, stream);
  cost_max_kernel<<<dim3(NPRIOR / 256, BATCH), 256, 0, stream>>>(Kmat, zz, pp,
                                                                 maxbits, NPRIOR);
  const long long total = (long long)BATCH * NPRIOR;
  expK_kernel<<<dim3((unsigned)((total + 255) / 256)), 256, 0, stream>>>(Kmat, maxbits,
                                                                         total);

  // ---- Sinkhorn: K (and K^T, if ws allows) stay resident in the 192MB L2 ----
  if (useKT) {
    transpose_kernel<<<dim3(NPRIOR / 32, BATCH / 32), 256, 0, stream>>>(Kmat, KT, NPRIOR);
    for (int it = 0; it < SINK_ITERS; ++it) {
      rowdot_kernel<<<dim3(NPRIOR), 256, 0, stream>>>(KT, u, v, BATCH,
                                                      1.0f / (float)NPRIOR);
      rowdot_kernel<<<dim3(BATCH), 256, 0, stream>>>(Kmat, v, u, NPRIOR,
                                                     1.0f / (float)BATCH);
    }
  } else {
    for (int it = 0; it < SINK_ITERS; ++it) {
      sink_col_part<<<dim3(NPRIOR / 256, COL_SPLIT), 256, 0, stream>>>(Kmat, u, part,
                                                                       BATCH, NPRIOR);
      sink_col_finish<<<dim3(NPRIOR / 256), 256, 0, stream>>>(part, v, NPRIOR,
                                                              1.0f / (float)NPRIOR);
      rowdot_kernel<<<dim3(BATCH), 256, 0, stream>>>(Kmat, v, u, NPRIOR,
                                                     1.0f / (float)BATCH);
    }
  }

  // ---- pi = u * K * v ; z_selected = pi @ z_prior ----
  scale_pi_kernel<<<dim3(NPRIOR / 256, BATCH), 256, 0, stream>>>(Kmat, u, v, NPRIOR);
  launch_gemm(Kmat, z_prior, nullptr, actA, BATCH, DLAT, NPRIOR, false, false, stream);

  // ---- Decoder: 128 -> 512 -> 1024 -> 512 -> 1024 ----
  launch_gemm(actA, dw[0], db[0], actB, BATCH, 512,  DLAT, true,  false, stream);
  launch_gemm(actB, dw[1], db[1], actA, BATCH, 1024, 512,  true,  false, stream);
  launch_gemm(actA, dw[2], db[2], actB, BATCH, 512,  1024, true,  false, stream);
  launch_gemm(actB, dw[3], db[3], out,  BATCH, DIN,  512,  false, false, stream);
}